// Tokenizer_21122649162300
// MI455X (gfx1250) — compile-verified
//
#include <hip/hip_runtime.h>
#include <hip/hip_bf16.h>

// CDNA5 / gfx1250, wave32. WMMA f32 16x16x4 used as an exact fp32 16-way
// row-sum engine (A = ones, B = data with N = problem lane), so each of the
// 16 scan problems carried by a wave gets its logsumexp denominator and its
// entropy accumulator without any shuffle-reduction tree.

typedef __attribute__((ext_vector_type(2))) float v2f;
typedef __attribute__((ext_vector_type(8))) float v8f;

#define NEGC (-1000000000.0f)

constexpr int Bq = 64;
constexpr int Mq = 16;
constexpr int Lq = 128;

// Sum of 16 fp32 values per problem (problem = lane&15; lane pair (l, l^16)
// supplies 8 values each as 4 chunks of K=4). Result broadcast to both lanes.
// D = sum_c A_ones(16x4) x B_c(4x16): every element of column n of D equals
// the total sum for problem n, independent of the hardware K-slot layout.
__device__ __forceinline__ float rowsum16_wmma(const float (&e)[4][2]) {
  v8f acc = {0.f, 0.f, 0.f, 0.f, 0.f, 0.f, 0.f, 0.f};
  const v2f ones = {1.0f, 1.0f};
#pragma unroll
  for (int c = 0; c < 4; ++c) {
    v2f bvec;
    bvec.x = e[c][0];
    bvec.y = e[c][1];
    acc = __builtin_amdgcn_wmma_f32_16x16x4_f32(false, ones, false, bvec,
                                                (short)0, acc, false, false);
  }
  return acc[0];
}

// Window shift w' = [inject, w[:-1]] over the split lane-pair layout:
// lane half=0 holds m = {4c, 4c+1}, half=1 holds m = {4c+2, 4c+3}.
__device__ __forceinline__ void shift_window(float (&w)[4][2], float inject,
                                             int half) {
  float py[4];
#pragma unroll
  for (int c = 0; c < 4; ++c) py[c] = __shfl_xor(w[c][1], 16, 32);
  float prev = inject;  // value entering lo-lane slot m=0
#pragma unroll
  for (int c = 0; c < 4; ++c) {
    float nx = half ? py[c] : prev;  // lo: m=4c <- m=4c-1 ; hi: m=4c+2 <- 4c+1
    w[c][1] = w[c][0];               // m = ...+1 <- m = ...
    w[c][0] = nx;
    prev = py[c];                    // partner's old m=4c+3 feeds next chunk
  }
}

// ---------------------------------------------------------------------------
// Fused forward-algorithm + entropy over the 64 primary sequences.
// 4 waves, 16 problems (batch rows) per wave.
// ---------------------------------------------------------------------------
__global__ __launch_bounds__(128) void fwd_entropy_kernel(
    const float* __restrict__ ts, const float* __restrict__ ms,
    const int* __restrict__ lengths, float* __restrict__ out_la,
    float* __restrict__ out_ela, float* __restrict__ out_ent,
    float* __restrict__ out_eent) {
  const int tid = threadIdx.x;
  const int lane = tid & 31;
  const int wid = tid >> 5;          // 0..3
  const int half = (lane >> 4) & 1;  // which half of the 16-m window
  const int p = lane & 15;
  const int b = wid * 16 + p;  // batch row, 0..63
  const int len = lengths[b];

  __builtin_prefetch(ts + (unsigned)b * Mq * Lq, 0, 1);
  __builtin_prefetch(ms + (unsigned)b * Mq * Lq, 0, 1);

  // 32-bit per-slot base offsets (hoisted): row (b, m) start in (B,M,L)
  unsigned rowOff[4][2];
#pragma unroll
  for (int c = 0; c < 4; ++c)
#pragma unroll
    for (int e = 0; e < 2; ++e) {
      const int m = 4 * c + 2 * half + e;
      rowOff[c][e] = ((unsigned)b * Mq + m) * Lq;
    }

  float wla[4][2], wh[4][2];
#pragma unroll
  for (int c = 0; c < 4; ++c) {
    wla[c][0] = 0.f; wla[c][1] = 0.f;
    wh[c][0] = 0.f;  wh[c][1] = 0.f;
  }
  float selLA = 0.f, selH = 0.f;  // la_all[len], h_all[len] (len==0 -> 0)

  for (int j = 0; j < Lq; ++j) {
    float cand[4][2], mv[4][2];
    float mloc = -3.402823466e38f;
#pragma unroll
    for (int c = 0; c < 4; ++c) {
#pragma unroll
      for (int e = 0; e < 2; ++e) {
        const unsigned off = rowOff[c][e] + j;
        const float tv = ts[off];
        const float mk = ms[off];
        mv[c][e] = mk;
        const float base = tv + (1.0f - mk) * NEGC;
        const float cd = base + mk * wla[c][e];
        cand[c][e] = cd;
        mloc = fmaxf(mloc, cd);
      }
    }
    const float mx = fmaxf(mloc, __shfl_xor(mloc, 16, 32));

    float ex[4][2];
#pragma unroll
    for (int c = 0; c < 4; ++c)
#pragma unroll
      for (int e = 0; e < 2; ++e) ex[c][e] = expf(cand[c][e] - mx);
    const float la = logf(rowsum16_wmma(ex)) + mx;

    // entropy contributions: q*(w_h + la - cand) where mask>0
    float contrib[4][2];
#pragma unroll
    for (int c = 0; c < 4; ++c) {
#pragma unroll
      for (int e = 0; e < 2; ++e) {
        const bool valid = mv[c][e] > 0.0f;
        const float q = valid ? expf(cand[c][e] - la) : 0.0f;
        contrib[c][e] = valid ? q * (wh[c][e] + la - cand[c][e]) : 0.0f;
      }
    }
    const float h = rowsum16_wmma(contrib);

    // edge outputs (B, M, L), 32-bit offsets
#pragma unroll
    for (int c = 0; c < 4; ++c) {
#pragma unroll
      for (int e = 0; e < 2; ++e) {
        const unsigned off = rowOff[c][e] + j;
        out_ela[off] = cand[c][e];
        out_eent[off] = contrib[c][e];
      }
    }

    selLA = (j == len - 1) ? la : selLA;
    selH = (j == len - 1) ? h : selH;

    shift_window(wla, la, half);
    shift_window(wh, h, half);
  }

  if (lane < 16) {  // one writer per problem (after all WMMAs)
    out_la[b] = selLA;
    out_ent[b] = selH;
  }
}

// ---------------------------------------------------------------------------
// Backward: 8192 independent scans. One block per batch row b (8 waves x 16
// positions = all 128 positions). bwd_ts/ms staged once in LDS; the big
// (L,M,L) masks are evaluated arithmetically:
//   iq = L-1-i ; emask = (j >= iq+m) ; mmask = emask || (m==0 && j<iq)
// ---------------------------------------------------------------------------
__global__ __launch_bounds__(256) void bwd_kernel(
    const float* __restrict__ bts, const float* __restrict__ bms,
    const int* __restrict__ lengths, float* __restrict__ out_lb,
    float* __restrict__ out_elb) {
  __shared__ float s_ts[Mq * Lq];
  __shared__ float s_ms[Mq * Lq];

  const int b = blockIdx.x;
  const int tid = threadIdx.x;
  __builtin_amdgcn_s_wait_tensorcnt(0);  // no TDM in flight: free no-op

  for (int idx = tid; idx < Mq * Lq; idx += 256) {
    s_ts[idx] = bts[(unsigned)b * Mq * Lq + idx];
    s_ms[idx] = bms[(unsigned)b * Mq * Lq + idx];
  }
  __syncthreads();

  const int lane = tid & 31;
  const int wid = tid >> 5;  // 0..7
  const int half = (lane >> 4) & 1;
  const int p = lane & 15;
  const int i = wid * 16 + p;  // backward position 0..127
  const int iq = Lq - 1 - i;
  const int len = lengths[b];

  // 32-bit bases hoisted out of the scan loop.
  unsigned ldsOff[4][2];   // m*Lq into s_ts/s_ms
  unsigned elbOff[4][2];   // ((b*Lq+i)*Mq + m)*Lq into out_elb (< 2^31)
  int mThr[4][2];          // emask threshold iq + m
#pragma unroll
  for (int c = 0; c < 4; ++c)
#pragma unroll
    for (int e = 0; e < 2; ++e) {
      const int m = 4 * c + 2 * half + e;
      ldsOff[c][e] = (unsigned)m * Lq;
      elbOff[c][e] = (((unsigned)b * Lq + i) * Mq + m) * Lq;
      mThr[c][e] = iq + m;
    }

  float w[4][2];
#pragma unroll
  for (int c = 0; c < 4; ++c) { w[c][0] = 0.f; w[c][1] = 0.f; }
  float sel = 0.f;

  for (int j = 0; j < Lq; ++j) {
    float cand[4][2];
    float mloc = -3.402823466e38f;
#pragma unroll
    for (int c = 0; c < 4; ++c) {
#pragma unroll
      for (int e = 0; e < 2; ++e) {
        const float t = s_ts[ldsOff[c][e] + j];  // bcast across half-wave
        const float mk = s_ms[ldsOff[c][e] + j];
        const bool emb = (j >= mThr[c][e]);
        const bool isM0 = (c == 0) && (e == 0) && (half == 0);
        const bool mmb = emb || (isM0 && (j < iq));
        const float m2 = mmb ? mk : 0.0f;
        const float t2 = emb ? t : 0.0f;
        const float base = t2 + (1.0f - m2) * NEGC;
        const float cd = base + m2 * w[c][e];
        cand[c][e] = cd;
        mloc = fmaxf(mloc, cd);
        // elb = cand*emask + NEG*(1-emask)
        out_elb[elbOff[c][e] + j] = emb ? cd : NEGC;
      }
    }
    const float mx = fmaxf(mloc, __shfl_xor(mloc, 16, 32));

    float ex[4][2];
#pragma unroll
    for (int c = 0; c < 4; ++c)
#pragma unroll
      for (int e = 0; e < 2; ++e) ex[c][e] = expf(cand[c][e] - mx);
    const float la = logf(rowsum16_wmma(ex)) + mx;

    sel = (j == len - 1) ? la : sel;
    shift_window(w, la, half);
  }

  if (lane < 16) out_lb[(unsigned)b * Lq + i] = sel;
}

// ---------------------------------------------------------------------------
// Host entry. Output layout (fp32, concatenated in reference return order):
//   fwd_ts[131072] fwd_ms[131072] log_alpha[64] edge_log_alpha[131072]
//   log_betas[8192] elb[16777216] ent[64] edge_ent[131072]
// ---------------------------------------------------------------------------
extern "C" void kernel_launch(void* const* d_in, const int* in_sizes, int n_in,
                              void* d_out, int out_size, void* d_ws,
                              size_t ws_size, hipStream_t stream) {
  (void)in_sizes; (void)n_in; (void)out_size; (void)d_ws; (void)ws_size;

  const float* fwd_ts = (const float*)d_in[0];
  const float* fwd_ms = (const float*)d_in[1];
  const float* bwd_ts = (const float*)d_in[2];
  const float* bwd_ms = (const float*)d_in[3];
  const int* lengths = (const int*)d_in[4];
  // d_in[5] = override_M == 16 == Mq (baked into the kernels)

  float* out = (float*)d_out;
  const size_t nBM_L = (size_t)Bq * Mq * Lq;  // 131072
  float* o_fts = out;
  float* o_fms = o_fts + nBM_L;
  float* o_la = o_fms + nBM_L;
  float* o_ela = o_la + Bq;
  float* o_lb = o_ela + nBM_L;
  float* o_elb = o_lb + (size_t)Bq * Lq;
  float* o_ent = o_elb + (size_t)Bq * Lq * Mq * Lq;
  float* o_eent = o_ent + Bq;

  hipMemcpyAsync(o_fts, fwd_ts, nBM_L * sizeof(float),
                 hipMemcpyDeviceToDevice, stream);
  hipMemcpyAsync(o_fms, fwd_ms, nBM_L * sizeof(float),
                 hipMemcpyDeviceToDevice, stream);

  fwd_entropy_kernel<<<1, 128, 0, stream>>>(fwd_ts, fwd_ms, lengths, o_la,
                                            o_ela, o_ent, o_eent);
  bwd_kernel<<<Bq, 256, 0, stream>>>(bwd_ts, bwd_ms, lengths, o_lb, o_elb);
}